// GravNetGNN_39608188403839
// MI455X (gfx1250) — compile-verified
//
#include <hip/hip_runtime.h>

typedef float v2f __attribute__((ext_vector_type(2)));
typedef float v8f __attribute__((ext_vector_type(8)));

#define D_FEAT 23
#define S_DIM 3
#define P_DIM 3
#define K_NN 16

// ---------------------------------------------------------------------------
// Kernel A: per-point projections s = xW_s + b_s, h = xW_h + b_h, sq = |s|^2
// s stored padded to float4 (w = 0) so it drops straight into the K=4 WMMA.
// ---------------------------------------------------------------------------
__global__ void gravnet_proj(const float* __restrict__ x,
                             const float* __restrict__ W_s, const float* __restrict__ b_s,
                             const float* __restrict__ W_h, const float* __restrict__ b_h,
                             float4* __restrict__ s4, float4* __restrict__ h4,
                             float* __restrict__ sq, int N) {
  int i = blockIdx.x * blockDim.x + threadIdx.x;
  if (i >= N) return;
  const float* xr = x + (size_t)i * D_FEAT;
  float s0 = b_s[0], s1 = b_s[1], s2 = b_s[2];
  float h0 = b_h[0], h1 = b_h[1], h2 = b_h[2];
#pragma unroll
  for (int c = 0; c < D_FEAT; ++c) {
    float xv = xr[c];
    s0 = fmaf(xv, W_s[c * S_DIM + 0], s0);
    s1 = fmaf(xv, W_s[c * S_DIM + 1], s1);
    s2 = fmaf(xv, W_s[c * S_DIM + 2], s2);
    h0 = fmaf(xv, W_h[c * P_DIM + 0], h0);
    h1 = fmaf(xv, W_h[c * P_DIM + 1], h1);
    h2 = fmaf(xv, W_h[c * P_DIM + 2], h2);
  }
  s4[i] = make_float4(s0, s1, s2, 0.0f);
  h4[i] = make_float4(h0, h1, h2, 0.0f);
  sq[i] = s0 * s0 + s1 * s1 + s2 * s2;
}

// ---------------------------------------------------------------------------
// Kernel B: one wave32 per 16-row stripe.
//  - WMMA f32 16x16x4 computes 16x16 Gram tiles of s @ s.T   (L2 resident)
//  - running per-row top-16 (split across lane pairs t / t+16)
//  - kNN aggregation + output GEMM + beta, all in the same wave.
// ---------------------------------------------------------------------------
__global__ __launch_bounds__(32)
void gravnet_knn(const float4* __restrict__ s4, const float4* __restrict__ h4,
                 const float* __restrict__ sq, const float* __restrict__ x,
                 const float* __restrict__ W_o1, const float* __restrict__ W_o2,
                 const float* __restrict__ b_o2, const float* __restrict__ W_beta,
                 const float* __restrict__ b_beta,
                 float* __restrict__ out_beta, float* __restrict__ out_latent,
                 int N) {
  __shared__ float tile[16][17];   // 16x16 d2 tile (+pad vs bank conflicts)
  __shared__ float mdist[32][16];  // per-lane partial top-K lists for merge
  __shared__ int   midx[32][16];

  const int lane = threadIdx.x;  // 0..31
  const int half = lane >> 4;    // 0: K 0/1 + rows 0..7 ; 1: K 2/3 + rows 8..15
  const int l16  = lane & 15;
  const int i0   = blockIdx.x * 16;

  // ---- A operand: rows i0..i0+15, 16x4 f32 layout (2 VGPRs / lane) ----
  float4 sv = s4[i0 + l16];
  v2f a;
  a[0] = half ? sv.z : sv.x;
  a[1] = half ? sv.w : sv.y;

  // sq for the 8 output rows this lane's C registers correspond to
  float sqi[8];
#pragma unroll
  for (int r = 0; r < 8; ++r) sqi[r] = sq[i0 + half * 8 + r];

  // top-K list, sorted ascending (dist[15] = current worst)
  float dist[16];
  int   idx[16];
#pragma unroll
  for (int k = 0; k < 16; ++k) { dist[k] = __builtin_inff(); idx[k] = 0; }

  const int ntiles = N >> 4;
  for (int jt = 0; jt < ntiles; ++jt) {
    const int j0 = jt << 4;

    // ---- B operand: columns j0..j0+15 of s.T (same per-lane packing) ----
    float4 sj = s4[j0 + l16];
    v2f b;
    b[0] = half ? sj.z : sj.x;
    b[1] = half ? sj.w : sj.y;
    float sqj = sq[j0 + l16];

    if (jt + 1 < ntiles) __builtin_prefetch(&s4[j0 + 16 + l16], 0, 1);

    v8f c = {0.f, 0.f, 0.f, 0.f, 0.f, 0.f, 0.f, 0.f};
    c = __builtin_amdgcn_wmma_f32_16x16x4_f32(false, a, false, b,
                                              (short)0, c, false, false);

    __syncthreads();  // previous tile's readers are done
#pragma unroll
    for (int r = 0; r < 8; ++r) {
      // d2[i][j] = |s_i|^2 + |s_j|^2 - 2 * <s_i, s_j>
      tile[half * 8 + r][l16] = sqi[r] + sqj - 2.0f * c[r];
    }
    __syncthreads();

    // ---- selection: lane owns row l16, columns half*8 .. half*8+7 ----
    float cand[8];
#pragma unroll
    for (int q = 0; q < 8; ++q) cand[q] = tile[l16][half * 8 + q];
    float m = cand[0];
#pragma unroll
    for (int q = 1; q < 8; ++q) m = fminf(m, cand[q]);

    if (m < dist[15]) {  // wave-coherent skip for late tiles
#pragma unroll
      for (int q = 0; q < 8; ++q) {
        float dv = cand[q];
        int   iv = j0 + half * 8 + q;
#pragma unroll
        for (int k = 0; k < 16; ++k) {
          if (dv < dist[k]) {
            float td = dist[k]; dist[k] = dv; dv = td;
            int   ti = idx[k];  idx[k]  = iv; iv = ti;
          }
        }
      }
    }
  }

  // ---- merge the two half-lists per row, then finish the row ----
  __syncthreads();
#pragma unroll
  for (int k = 0; k < 16; ++k) { mdist[lane][k] = dist[k]; midx[lane][k] = idx[k]; }
  __syncthreads();

  if (lane < 16) {
    const int row = i0 + lane;
    int pa = 0, pb = 0;
    float fd[16];
    int   fi[16];
#pragma unroll
    for (int k = 0; k < 16; ++k) {  // pa+pb == k <= 15, so reads stay in range
      float da = mdist[lane][pa];
      float db = mdist[lane + 16][pb];
      if (da <= db) { fd[k] = da; fi[k] = midx[lane][pa];      ++pa; }
      else          { fd[k] = db; fi[k] = midx[lane + 16][pb]; ++pb; }
    }

    // w = exp(-10 * max(d2, 0)); msgs = h[idx] * w; mean & max over K
    float sum0 = 0.f, sum1 = 0.f, sum2 = 0.f;
    float mx0 = -__builtin_inff(), mx1 = -__builtin_inff(), mx2 = -__builtin_inff();
#pragma unroll
    for (int k = 0; k < K_NN; ++k) {
      float w = __expf(-10.0f * fmaxf(fd[k], 0.0f));
      float4 hv = h4[fi[k]];
      float m0 = hv.x * w, m1 = hv.y * w, m2 = hv.z * w;
      sum0 += m0; sum1 += m1; sum2 += m2;
      mx0 = fmaxf(mx0, m0); mx1 = fmaxf(mx1, m1); mx2 = fmaxf(mx2, m2);
    }
    const float inv_k = 1.0f / (float)K_NN;
    float aggr[2 * P_DIM] = {sum0 * inv_k, sum1 * inv_k, sum2 * inv_k, mx0, mx1, mx2};

    // latent = x @ W_o1 + aggr @ W_o2 + b_o2
    float xv[D_FEAT];
    const float* xr = x + (size_t)row * D_FEAT;
#pragma unroll
    for (int cc = 0; cc < D_FEAT; ++cc) xv[cc] = xr[cc];

    float lat[D_FEAT];
#pragma unroll
    for (int dcol = 0; dcol < D_FEAT; ++dcol) {
      float acc = b_o2[dcol];
#pragma unroll
      for (int cc = 0; cc < D_FEAT; ++cc) acc = fmaf(xv[cc], W_o1[cc * D_FEAT + dcol], acc);
#pragma unroll
      for (int p = 0; p < 2 * P_DIM; ++p) acc = fmaf(aggr[p], W_o2[p * D_FEAT + dcol], acc);
      lat[dcol] = acc;
    }

    // beta = clip(sigmoid(latent @ W_beta + b_beta), eps, 1-eps)
    float z = b_beta[0];
#pragma unroll
    for (int dcol = 0; dcol < D_FEAT; ++dcol) z = fmaf(lat[dcol], W_beta[dcol], z);
    float beta = 1.0f / (1.0f + __expf(-z));
    beta = fminf(fmaxf(beta, 1e-6f), 1.0f - 1e-6f);

    out_beta[row] = beta;
#pragma unroll
    for (int dcol = 0; dcol < D_FEAT; ++dcol) out_latent[(size_t)row * D_FEAT + dcol] = lat[dcol];
  }
}

// ---------------------------------------------------------------------------
extern "C" void kernel_launch(void* const* d_in, const int* in_sizes, int n_in,
                              void* d_out, int out_size, void* d_ws, size_t ws_size,
                              hipStream_t stream) {
  const float* x      = (const float*)d_in[0];
  const float* W_s    = (const float*)d_in[1];
  const float* b_s    = (const float*)d_in[2];
  const float* W_h    = (const float*)d_in[3];
  const float* b_h    = (const float*)d_in[4];
  const float* W_o1   = (const float*)d_in[5];
  const float* W_o2   = (const float*)d_in[6];
  const float* b_o2   = (const float*)d_in[7];
  const float* W_beta = (const float*)d_in[8];
  const float* b_beta = (const float*)d_in[9];

  const int N = in_sizes[0] / D_FEAT;  // 16384, multiple of 16

  char* ws = (char*)d_ws;
  float4* s4 = (float4*)ws; ws += (size_t)N * sizeof(float4);
  float4* h4 = (float4*)ws; ws += (size_t)N * sizeof(float4);
  float*  sq = (float*)ws;

  float* out_beta   = (float*)d_out;   // tuple order: beta [N], then latent [N*D]
  float* out_latent = out_beta + N;

  gravnet_proj<<<(N + 255) / 256, 256, 0, stream>>>(x, W_s, b_s, W_h, b_h, s4, h4, sq, N);
  gravnet_knn<<<N / 16, 32, 0, stream>>>(s4, h4, sq, x, W_o1, W_o2, b_o2, W_beta, b_beta,
                                         out_beta, out_latent, N);
}